// RotatEScore_1872605741814
// MI455X (gfx1250) — compile-verified
//
#include <hip/hip_runtime.h>
#include <hip/hip_bf16.h>
#include <cstdint>
#include <math.h>

// RotatE negative-sampling score for MI455X (gfx1250).
//
// Roofline: 134M inner iterations (each ~10 VALU + 2 v_sqrt_f32 per k for 2
// rows) vs ~10 MB of traffic -> transcendental/VALU bound, fully L2-resident.
// The per-element sqrt prevents WMMA factorization, so the CDNA5-specific
// path is the Tensor Data Mover: one tensor_load_to_lds per block DMAs the
// 32x512 f32 tails tile into LDS (TDM row padding -> conflict-free reads),
// overlapped with the prologue and synced with s_wait_tensorcnt.
//
// New this round: the sincos rotation (only depends on (chunk, i-row)) was
// ~25% of total VALU and recomputed once per j-tile (8x). It is now
// precomputed once into d_ws (4 MB of float2) by a tiny kernel; the main
// kernel does a coalesced ws->LDS copy instead. Falls back to the fused
// variant if ws_size is too small (host-side, deterministic).

typedef unsigned int u32;
typedef unsigned long long u64;
typedef __attribute__((ext_vector_type(4))) u32 v4u;
typedef __attribute__((ext_vector_type(4))) int v4i;
typedef __attribute__((ext_vector_type(8))) int v8i;

#define GAMMA_F 12.0f
#define PHASE_SCALE 62.83185307179586477f /* pi / 0.05 */

#define TI 16    /* head rows per block                 */
#define TJ 32    /* tail rows per block                 */
#define H 256    /* hidden_dim / 2                      */
#define HD 512   /* hidden_dim                          */
#define NROWS 2048 /* num_chunks * chunk_size           */
#define TPAD 514 /* padded LDS row stride in dwords: 512 + 2 pads (1 dw per 256 dw) */

__device__ __forceinline__ u32 lds_offset_u32(const void* p) {
  // generic -> LDS addrspace cast; AS(3) pointers are 32-bit LDS byte offsets
  return (u32)(uintptr_t)(__attribute__((address_space(3))) const void*)p;
}

// ---- kernel 1 (split mode): rotate heads once: rot[row*H+k] = (re', im') ----
__global__ __launch_bounds__(256) void rotate_heads_kernel(
    const float* __restrict__ heads,      // [2048, 512]
    const float* __restrict__ relations,  // [2048, 256]
    float2* __restrict__ rot)             // [2048, 256]
{
  const int row = blockIdx.x;   // 2048
  const int k = threadIdx.x;    // 256 (coalesced)
  const float re = heads[(size_t)row * HD + k];
  const float im = heads[(size_t)row * HD + H + k];
  const float ph = relations[(size_t)row * H + k] * PHASE_SCALE;
  const float sr = sinf(ph);
  const float cr = cosf(ph);
  float2 v;
  v.x = re * cr - im * sr;
  v.y = re * sr + im * cr;
  rot[(size_t)row * H + k] = v;
}

// ---- kernel 2: main score kernel -------------------------------------------
template <bool USE_ROT>
__global__ __launch_bounds__(256) void rotate_score_kernel(
    const float* __restrict__ heads,      // [2048, 512]
    const float* __restrict__ relations,  // [2048, 256]
    const float* __restrict__ tails,      // [1024, 512]
    const float2* __restrict__ rot,       // [2048, 256] or unused
    float* __restrict__ out)              // [4, 512, 256]
{
  __shared__ __align__(128) float sTails[TJ * TPAD];  // 65,792 B
  __shared__ __align__(16) float2 sHead[TI * H];      // 32,768 B

  const int tid = threadIdx.x;
  const int b = blockIdx.x;     // 1024 blocks
  const int c = b >> 8;         // chunk 0..3
  const int it = (b & 255) >> 3;// i-tile 0..31
  const int jt = b & 7;         // j-tile 0..7

  // ---------------- stage 1: kick off TDM DMA of the tails tile -------------
#if __has_builtin(__builtin_amdgcn_tensor_load_to_lds)
  if (tid < 32) {  // wave 0 only issues the DMA (TDM ignores EXEC)
    const u64 ga =
        (u64)(uintptr_t)(tails + (size_t)(c * 256 + jt * TJ) * HD);
    v4u g0;
    g0[0] = 1u;                         // count=1, user descriptor, no gather
    g0[1] = lds_offset_u32(&sTails[0]); // lds_addr
    g0[2] = (u32)ga;                    // global_addr[31:0]
    g0[3] = (u32)((ga >> 32) & 0x01FFFFFFu) | 0x80000000u; // addr[56:32] | type=2

    v8i g1;
    // data_size=2 (4B) | pad_enable | pad_interval=7 (every 256 dw) | pad_amount=0 (1 dw)
    g1[0] = (int)((2u << 16) | (1u << 20) | (7u << 22));
    g1[1] = (int)((u32)HD << 16); // tensor_dim0 = 512 elements / row
    g1[2] = (int)((u32)TJ << 16); // tensor_dim1 = 32 rows
    g1[3] = (int)((u32)HD << 16); // tile_dim0   = 512
    g1[4] = TJ;                   // tile_dim1   = 32
    g1[5] = HD;                   // tensor_dim0_stride = 512 elements
    g1[6] = 0;
    g1[7] = 0;

    v4i gz4 = {0, 0, 0, 0};            // groups 2/3 unused (2-D tile)
    v8i gz8 = {0, 0, 0, 0, 0, 0, 0, 0};
    __builtin_amdgcn_tensor_load_to_lds(g0, g1, gz4, gz4, gz8, 0);
  }
#else
  { // fallback: cooperative coalesced copy into the same padded layout
    const size_t base = (size_t)(c * 256 + jt * TJ) * HD;
    for (int idx = tid; idx < TJ * HD; idx += 256) {
      const int row = idx >> 9, col = idx & (HD - 1);
      sTails[row * TPAD + col + (col >= H ? 1 : 0)] = tails[base + idx];
    }
  }
#endif

  // ------- stage 2 (overlaps the DMA): rotated heads into LDS ---------------
  const int row0 = c * 512 + it * TI;
  if (USE_ROT) {
    // precomputed rotation: plain coalesced copy (16x256 float2, contiguous)
    const float2* __restrict__ src = rot + (size_t)row0 * H;
#pragma unroll
    for (int idx = tid; idx < TI * H; idx += 256) sHead[idx] = src[idx];
  } else {
    const int k = tid;  // 0..255, coalesced across the block
#pragma unroll 4
    for (int i = 0; i < TI; ++i) {
      const int gi = row0 + i;
      const float re = heads[(size_t)gi * HD + k];
      const float im = heads[(size_t)gi * HD + H + k];
      const float ph = relations[(size_t)gi * H + k] * PHASE_SCALE;
      const float sr = sinf(ph);
      const float cr = cosf(ph);
      float2 v;
      v.x = re * cr - im * sr;
      v.y = re * sr + im * cr;
      sHead[i * H + k] = v;  // ds_store_b64
    }
  }

#if __has_builtin(__builtin_amdgcn_s_wait_tensorcnt)
  __builtin_amdgcn_s_wait_tensorcnt(0);  // no-op for waves with TENSORcnt==0
#endif
  __syncthreads();

  // ---------------- stage 3: hot loop (VALU / v_sqrt bound) -----------------
  const int jl = tid & 31;   // lane -> tail row within tile (conflict-free: 514 % 64 == 2)
  const int isub = tid >> 5; // wave id 0..7 -> uniform sHead broadcasts
  const float* __restrict__ tr = &sTails[jl * TPAD];       // t_real at dw k
  const float* __restrict__ ti = tr + (H + 1);             // t_imag at dw 257+k (1 pad dw)
  const float2* __restrict__ h0 = &sHead[isub * H];
  const float2* __restrict__ h1 = &sHead[(isub + 8) * H];

  float acc0 = 0.0f, acc1 = 0.0f;
#pragma unroll 4
  for (int k = 0; k < H; ++k) {
    const float t_r = tr[k];
    const float t_i = ti[k];
    float2 a = h0[k];
    float dr = a.x - t_r, di = a.y - t_i;
    acc0 += __builtin_amdgcn_sqrtf(dr * dr + di * di);
    float2 bb = h1[k];
    dr = bb.x - t_r; di = bb.y - t_i;
    acc1 += __builtin_amdgcn_sqrtf(dr * dr + di * di);
  }

  const int jg = jt * TJ + jl;
  const int i0 = it * TI + isub;
  out[(size_t)(c * 512 + i0) * 256 + jg] = GAMMA_F - acc0;
  out[(size_t)(c * 512 + i0 + 8) * 256 + jg] = GAMMA_F - acc1;
}

extern "C" void kernel_launch(void* const* d_in, const int* in_sizes, int n_in,
                              void* d_out, int out_size, void* d_ws, size_t ws_size,
                              hipStream_t stream) {
  (void)in_sizes; (void)n_in; (void)out_size;
  const float* heads = (const float*)d_in[0];
  const float* relations = (const float*)d_in[1];
  const float* tails = (const float*)d_in[2];
  float* out = (float*)d_out;

  // dims fixed by setup_inputs(): 4 chunks x 512 chunk_size x 256 negatives, hidden 512
  const size_t rot_bytes = (size_t)NROWS * H * sizeof(float2);  // 4 MB
  if (ws_size >= rot_bytes) {
    float2* rot = (float2*)d_ws;
    rotate_heads_kernel<<<dim3(NROWS), dim3(256), 0, stream>>>(heads, relations, rot);
    rotate_score_kernel<true><<<dim3(4 * 32 * 8), dim3(256), 0, stream>>>(
        heads, relations, tails, rot, out);
  } else {
    rotate_score_kernel<false><<<dim3(4 * 32 * 8), dim3(256), 0, stream>>>(
        heads, relations, tails, nullptr, out);
  }
}